// Pool4d_91104846283379
// MI455X (gfx1250) — compile-verified
//
#include <hip/hip_runtime.h>

// Pool4d: x[2,8,16,32,64,64] f32, window (3,3,3,3) over (l,d,h,w),
// strides (1,2,2,2), VALID padding, * 1/81 -> out[2,8,14,15,31,31].
// Memory-bound (134 MB in / 12.9 MB out @ 23.3 TB/s ~ 6.3 us floor).
// Strategy: per-WG LDS tile staged via CDNA5 async global->LDS loads,
// two-stage in-LDS reduction, coalesced stores. Whole input fits in the
// 192 MB L2, so tile halo re-reads are cheap.

#define B_  2
#define C_  8
#define L_  16
#define D_  32
#define H_  64
#define W_  64
#define LO_ 14
#define DO_ 15
#define HO_ 31
#define WO_ 31

#define NTHREADS 256

#if defined(__HIP_DEVICE_COMPILE__)
#if __has_builtin(__builtin_amdgcn_global_load_async_to_lds_b128)
#define HAVE_ASYNC_LDS 1
#endif
#endif

typedef int v4i_ __attribute__((vector_size(16)));
typedef __attribute__((address_space(1))) v4i_ gv4i;   // global int4
typedef __attribute__((address_space(3))) v4i_ lv4i;   // LDS int4

__global__ __launch_bounds__(NTHREADS)
void pool4d_kernel(const float* __restrict__ x, float* __restrict__ out) {
  // Input tile: [l=16][dd=3][hh=3][w=64] floats = 36 KB
  __shared__ float T[L_ * 3 * 3 * W_];
  // Stage-1 (3x3 h,w pooled) sums: [l=16][dd=3][wo padded to 32] = 6 KB
  __shared__ float S[L_ * 3 * 32];

  const int tid = threadIdx.x;
  int bid = blockIdx.x;
  const int ho = bid % HO_; bid /= HO_;   // output h index == h tile
  const int dO = bid % DO_; bid /= DO_;   // output d index
  const int c  = bid % C_;  bid /= C_;
  const int b  = bid;

  const int h0 = 2 * ho;                  // input h base (<= 60, +2 <= 62 < 64)
  const int d0 = 2 * dO;                  // input d base (<= 28, +2 <= 30 < 32)
  const int bc_base = ((b * C_ + c) * L_) * (D_ * H_ * W_);

  // ---------------- Load input tile into LDS (async b128 per lane) --------
  // 9216 floats = 2304 float4 = exactly 9 per thread at 256 threads.
  constexpr int NVEC = (L_ * 3 * 3 * W_) / 4;  // 2304
#pragma unroll
  for (int it = 0; it < NVEC / NTHREADS; ++it) {
    const int q   = tid + it * NTHREADS;
    const int w4  = q & 15;        // 16 float4 per 64-float row
    const int row = q >> 4;        // [0,144)
    const int hh  = row % 3;
    const int dd  = (row / 3) % 3;
    const int l   = row / 9;
    const int goff = bc_base +
        ((l * D_ + (d0 + dd)) * H_ + (h0 + hh)) * W_ + (w4 << 2);
    float* ldst = &T[q << 2];      // LDS offset = q*16 bytes (contiguous)
#if defined(HAVE_ASYNC_LDS)
    __builtin_amdgcn_global_load_async_to_lds_b128(
        (gv4i*)(x + goff),
        (lv4i*)ldst,
        /*offset=*/0, /*cpol=*/0);
#else
    *(float4*)ldst = *(const float4*)(x + goff);
#endif
  }

#if defined(HAVE_ASYNC_LDS)
#if __has_builtin(__builtin_amdgcn_s_wait_asynccnt)
  __builtin_amdgcn_s_wait_asynccnt(0);
#else
  asm volatile("s_wait_asynccnt 0" ::: "memory");
#endif
#endif
  __syncthreads();

  // ---------------- Stage 1: 3x3 pool over (h,w) --------------------------
  // S[l][dd][wo] = sum_{dh,dw in 0..2} T[l][dd][dh][2*wo+dw]
  constexpr int NS = L_ * 3 * WO_;  // 1488
  for (int e = tid; e < NS; e += NTHREADS) {
    const int wo = e % WO_;
    const int t  = e / WO_;        // l*3 + dd
    const float* tp = &T[(t * 3) * W_ + 2 * wo];
    float s = 0.f;
#pragma unroll
    for (int dh = 0; dh < 3; ++dh) {
      const float* r = tp + dh * W_;
      s += r[0] + r[1] + r[2];
    }
    S[t * 32 + wo] = s;
  }
  __syncthreads();

  // ---------------- Stage 2: 3x3 pool over (l,d), scale, store ------------
  constexpr int NO = LO_ * WO_;  // 434
  const int out_base = ((((b * C_ + c) * LO_) * DO_ + dO) * HO_ + ho) * WO_;
  const int out_lstride = DO_ * HO_ * WO_;
  for (int e = tid; e < NO; e += NTHREADS) {
    const int wo = e % WO_;
    const int lo = e / WO_;
    float s = 0.f;
#pragma unroll
    for (int dl = 0; dl < 3; ++dl) {
#pragma unroll
      for (int dd = 0; dd < 3; ++dd) {
        s += S[((lo + dl) * 3 + dd) * 32 + wo];
      }
    }
    out[out_base + lo * out_lstride + wo] = s * (1.0f / 81.0f);
  }
}

extern "C" void kernel_launch(void* const* d_in, const int* in_sizes, int n_in,
                              void* d_out, int out_size, void* d_ws, size_t ws_size,
                              hipStream_t stream) {
  (void)in_sizes; (void)n_in; (void)d_ws; (void)ws_size; (void)out_size;
  const float* x = (const float*)d_in[0];
  float* out = (float*)d_out;
  const int nblocks = B_ * C_ * DO_ * HO_;  // 7440
  pool4d_kernel<<<dim3(nblocks), dim3(NTHREADS), 0, stream>>>(x, out);
}